// Net_22746146799726
// MI455X (gfx1250) — compile-verified
//
#include <hip/hip_runtime.h>

#define D 64

typedef float v2f __attribute__((ext_vector_type(2)));
typedef float v8f __attribute__((ext_vector_type(8)));

// ---------------- degree / normalization ----------------

__global__ void k_deg_init(float* __restrict__ deg, int n) {
    int i = blockIdx.x * blockDim.x + threadIdx.x;
    if (i < n) deg[i] = 1.0f;   // self loop contributes 1 to every node
}

__global__ void k_deg_accum(const long long* __restrict__ dst, int e,
                            float* __restrict__ deg) {
    int i = blockIdx.x * blockDim.x + threadIdx.x;
    if (i < e) atomicAdd(&deg[(int)dst[i]], 1.0f);
}

__global__ void k_rsqrt_inplace(float* __restrict__ deg, int n) {
    int i = blockIdx.x * blockDim.x + threadIdx.x;
    if (i < n) deg[i] = rsqrtf(deg[i]);   // deg >= 1 always (self loops)
}

// ---------------- fused GEMM + self-loop/bias epilogue ----------------
// H = X @ W^T via V_WMMA_F32_16X16X4_F32 (exact fp32 GEMM).
// Epilogue also writes OutInit = H * dinv^2 + b, which seeds the scatter
// target with the self-loop term and the bias -- saves a separate 51 MB
// read+write sweep per layer.
//
// LDS holds W transposed and K-pair-interleaved:
//   Wp[k/2][n] = (W[n][k], W[n][k+1])  as float2
// so each B fragment is one aligned ds_load_b64 straight into the even
// VGPR pair the WMMA consumes (no v_mov shuffles).

__global__ __launch_bounds__(256) void k_gcn_gemm(const float* __restrict__ X,
                                                  const float* __restrict__ W,
                                                  const float* __restrict__ bias,
                                                  const float* __restrict__ dinv,
                                                  float* __restrict__ H,
                                                  float* __restrict__ OutInit,
                                                  int n) {
    __shared__ v2f Wp[32 * D];            // 16 KB
    float* Wf = (float*)Wp;
    int t = threadIdx.x;
    for (int i = t; i < D * D; i += 256) {
        int o = i >> 6, k = i & 63;       // W[o][k]
        Wf[((k >> 1) << 7) + (o << 1) + (k & 1)] = W[i];
    }
    __syncthreads();

    int wave = t >> 5;
    int lane = t & 31;
    int tile = blockIdx.x * 8 + wave;           // 16-row tile index
    if (tile * 16 >= n) return;                 // wave-uniform guard (EXEC all 1s)
    int rowBase = tile * 16;

    int m    = lane & 15;                       // M (A) / N (B,C,D) position
    int koff = (lane >> 4) << 1;                // 0 for lanes 0-15, 2 for 16-31

    const float* xr = X + (size_t)(rowBase + m) * D;

    v8f acc[4] = {};                            // 4 column tiles of 16
#pragma unroll
    for (int k0 = 0; k0 < D; k0 += 4) {
        int kk = k0 + koff;                     // even
        v2f a = *(const v2f*)(xr + kk);         // A frag: K=kk, kk+1 for row m
#pragma unroll
        for (int tn = 0; tn < 4; ++tn) {
            v2f b = Wp[((kk >> 1) << 6) + tn * 16 + m];  // (W[n][kk], W[n][kk+1])
            acc[tn] = __builtin_amdgcn_wmma_f32_16x16x4_f32(
                false, a, false, b, (short)0, acc[tn], false, false);
        }
    }

    // epilogue: H and self-loop+bias init of the scatter target
    float bb[4];
#pragma unroll
    for (int tn = 0; tn < 4; ++tn) bb[tn] = bias[tn * 16 + m];

    int rAdd = (lane >> 4) << 3;                // C/D: lanes 16-31 hold M=r+8
#pragma unroll
    for (int r = 0; r < 8; ++r) {
        int row = rowBase + r + rAdd;
        float di = dinv[row];
        float di2 = di * di;
#pragma unroll
        for (int tn = 0; tn < 4; ++tn) {
            float val = acc[tn][r];
            size_t idx = (size_t)row * D + tn * 16 + m;
            H[idx] = val;
            OutInit[idx] = val * di2 + bb[tn];
        }
    }
}

// ---------------- per-edge gather/scale/scatter-add -------------------------
// One wave per edge; each lane owns a float2 column pair. Gather is two
// coalesced 128B requests per edge; scatter is f32 global atomics (L2-resident:
// 25.6 MB output fits in the 192 MB L2).

__global__ void k_edge_agg(const long long* __restrict__ src,
                           const long long* __restrict__ dst,
                           const float* __restrict__ dinv,
                           const float* __restrict__ H,
                           float* __restrict__ out, int e) {
    int gid  = blockIdx.x * blockDim.x + threadIdx.x;
    int edge = gid >> 5;
    if (edge >= e) return;
    int lane = gid & 31;

    int s = (int)src[edge];
    int d = (int)dst[edge];
    float nr = dinv[s] * dinv[d];

    int c = lane * 2;
    v2f v = *(const v2f*)(H + (size_t)s * D + c);
    float* op = out + (size_t)d * D + c;
    atomicAdd(op,     v.x * nr);
    atomicAdd(op + 1, v.y * nr);
}

// ---------------- driver ----------------

extern "C" void kernel_launch(void* const* d_in, const int* in_sizes, int n_in,
                              void* d_out, int out_size, void* d_ws, size_t ws_size,
                              hipStream_t stream) {
    const float*     x   = (const float*)d_in[0];
    const long long* ei  = (const long long*)d_in[1];   // int64 [2, E]
    const float*     W1  = (const float*)d_in[2];
    const float*     b1  = (const float*)d_in[3];
    const float*     W2  = (const float*)d_in[4];
    const float*     b2  = (const float*)d_in[5];
    float*           out = (float*)d_out;

    int n = in_sizes[0] / D;        // 100000
    int e = in_sizes[1] / 2;        // 1600000
    const long long* srcp = ei;
    const long long* dstp = ei + e;

    float* ws   = (float*)d_ws;
    float* dinv = ws;                               // n floats
    float* h    = ws + 102400;                      // n*64 floats
    float* tmp  = h + (size_t)n * D;                // n*64 floats

    int thr = 256;
    int gN   = (n + thr - 1) / thr;
    int gE   = (e + thr - 1) / thr;
    int gEW  = (int)(((size_t)e * 32 + thr - 1) / thr);
    int tiles = (n + 15) / 16;
    int gG   = (tiles + 7) / 8;

    // symmetric normalization (shared by both layers)
    k_deg_init<<<gN, thr, 0, stream>>>(dinv, n);
    k_deg_accum<<<gE, thr, 0, stream>>>(dstp, e, dinv);
    k_rsqrt_inplace<<<gN, thr, 0, stream>>>(dinv, n);

    // layer 1: h = x@W1^T ; tmp seeded with self-loop+bias ; scatter edges
    k_gcn_gemm<<<gG, thr, 0, stream>>>(x, W1, b1, dinv, h, tmp, n);
    k_edge_agg<<<gEW, thr, 0, stream>>>(srcp, dstp, dinv, h, tmp, e);

    // layer 2: h = tmp@W2^T ; out seeded with self-loop+bias ; scatter edges
    k_gcn_gemm<<<gG, thr, 0, stream>>>(tmp, W2, b2, dinv, h, out, n);
    k_edge_agg<<<gEW, thr, 0, stream>>>(srcp, dstp, dinv, h, out, e);
}